// TwoPhaseBlockAttnRes_62328565400275
// MI455X (gfx1250) — compile-verified
//
#include <hip/hip_runtime.h>

// ---------------------------------------------------------------------------
// TwoPhaseBlockAttnRes for MI455X (gfx1250).
// Bandwidth-bound (~185 MB traffic @ 23.3 TB/s ≈ 8 us): read V once via
// async global->LDS b128 staging, logits via v_wmma_f32_16x16x4_f32 GEMM
// ([9x1024]x[1024x2] padded to 16x16 tiles), rest VALU from LDS.
// ---------------------------------------------------------------------------

#define D_DIM 1024
#define N_FULL 8
#define N_TOT 9           // 8 blocks + partial
#define B_DIM 2
#define T_DIM 2048
#define ROW_STRIDE 1032   // padded row stride in floats (16B-aligned rows)
#define EPS 1e-6f

typedef __attribute__((ext_vector_type(2))) float v2f;
typedef __attribute__((ext_vector_type(8))) float v8f;
typedef __attribute__((ext_vector_type(4))) int   v4i;

typedef v4i __attribute__((address_space(1)))* gptr_v4i;   // global int4*
typedef v4i __attribute__((address_space(3)))* lptr_v4i;   // LDS int4*

#if defined(__gfx1250__) && __has_builtin(__builtin_amdgcn_global_load_async_to_lds_b128)
#define HAVE_ASYNC 1
#else
#define HAVE_ASYNC 0
#endif

__global__ __launch_bounds__(256) void
block_attn_kernel(const float* __restrict__ blocks,
                  const float* __restrict__ partial,
                  const float* __restrict__ pqa,
                  const float* __restrict__ pqm,
                  const float* __restrict__ nwa,
                  const float* __restrict__ nwm,
                  float* __restrict__ out)
{
    __shared__ float xs[N_TOT * ROW_STRIDE];   // staged V rows
    __shared__ float qaw[D_DIM];               // pseudo_query_attn * norm_attn_w
    __shared__ float qmw[D_DIM];               // pseudo_query_mlp  * norm_mlp_w
    __shared__ float s_dot_a[16];
    __shared__ float s_dot_m[16];
    __shared__ float s_ss[N_TOT];
    __shared__ float s_wa[N_TOT];
    __shared__ float s_wm[N_TOT];

    const int tid  = threadIdx.x;
    const int wave = tid >> 5;
    const int lane = tid & 31;
    const int bt   = blockIdx.x;          // 0 .. B*T-1
    const int b    = bt / T_DIM;
    const int t    = bt % T_DIM;

    if (tid < 16) { s_dot_a[tid] = 0.0f; s_dot_m[tid] = 0.0f; }

    // Effective per-head query vectors (dot(q, rmsnorm(x,w)) == rms*dot(q*w, x))
    for (int i = tid; i < D_DIM; i += 256) {
        qaw[i] = pqa[i] * nwa[i];
        qmw[i] = pqm[i] * nwm[i];
    }

    // ---- Stage the 9 V rows for this (b,t) into LDS: 1 b128 per thread per row,
    // all 9 async DMAs in flight before a single ASYNCcnt drain.
    const float* psrc = partial + ((size_t)b * T_DIM + t) * (size_t)D_DIM;
#pragma unroll
    for (int n = 0; n < N_TOT; ++n) {
        const float* src = (n < N_FULL)
            ? blocks + ((((size_t)n * B_DIM + b) * T_DIM) + t) * (size_t)D_DIM
            : psrc;
#if HAVE_ASYNC
        __builtin_amdgcn_global_load_async_to_lds_b128(
            (gptr_v4i)(src + tid * 4),
            (lptr_v4i)(&xs[n * ROW_STRIDE + tid * 4]),
            0, 0);
#else
        float4 v = *(const float4*)(src + tid * 4);
        *(float4*)(&xs[n * ROW_STRIDE + tid * 4]) = v;
#endif
    }
#if HAVE_ASYNC
#if __has_builtin(__builtin_amdgcn_s_wait_asynccnt)
    __builtin_amdgcn_s_wait_asynccnt(0);
#else
    asm volatile("s_wait_asynccnt 0" ::: "memory");
#endif
#endif
    __syncthreads();

    // ---- Sum-of-squares per row (waves 0..7 cover rows, wave0 also row 8)
    // Lane-contiguous float4 reads: 8 conflict-free ds_load_b128 per row.
    for (int n = wave; n < N_TOT; n += 8) {
        const float* row = &xs[n * ROW_STRIDE];
        float ss = 0.0f;
#pragma unroll
        for (int j = 0; j < 8; ++j) {
            const float4 x4 = *(const float4*)(&row[j * 128 + lane * 4]);
            ss = fmaf(x4.x, x4.x, ss);
            ss = fmaf(x4.y, x4.y, ss);
            ss = fmaf(x4.z, x4.z, ss);
            ss = fmaf(x4.w, x4.w, ss);
        }
        for (int off = 16; off > 0; off >>= 1) ss += __shfl_xor(ss, off, 32);
        if (lane == 0) s_ss[n] = ss;
    }

    // ---- Logits GEMM via WMMA f32 16x16x4.
    // A: 16 rows (rows 0..8 = V rows, 9..15 ignored) x K=1024, wave w owns
    // K-slice [128w, 128w+128) as 32 chained wmmas. B: col0 = qaw, col1 = qmw.
    // A layout: lane L -> row M=L&15; K = k0 + 2*(L>>4) + vgpr.
    // B built branch-free: unconditional ds_load_b64 of both queries + fma blend
    // with per-lane 0/1 masks (avoids divergent exec-managed LDS loads).
    const int col  = lane & 15;
    const int half = lane >> 4;
    const int rA   = (col < N_TOT) ? col : 0;   // clamp pad rows into valid LDS
    const float* arow = &xs[rA * ROW_STRIDE];
    const float m_a = (col == 0) ? 1.0f : 0.0f;
    const float m_m = (col == 1) ? 1.0f : 0.0f;

    v8f acc = {};
    for (int kc = 0; kc < 32; ++kc) {
        const int kk = wave * 128 + kc * 4 + half * 2;
        const v2f a   = *(const v2f*)(&arow[kk]);   // 8B aligned (kk even)
        const v2f qa2 = *(const v2f*)(&qaw[kk]);
        const v2f qm2 = *(const v2f*)(&qmw[kk]);
        v2f bmat;
        bmat.x = fmaf(qa2.x, m_a, qm2.x * m_m);
        bmat.y = fmaf(qa2.y, m_a, qm2.y * m_m);
        acc = __builtin_amdgcn_wmma_f32_16x16x4_f32(
            /*neg_a=*/false, a, /*neg_b=*/false, bmat,
            /*c_mod=*/(short)0, acc, /*reuse_a=*/false, /*reuse_b=*/false);
    }

    // C layout: VGPR v, lane L -> M = v + 8*(L>>4), N = L&15.
    // Cross-wave K reduction of the partial dots via LDS float atomics.
#pragma unroll
    for (int v = 0; v < 8; ++v) {
        const float cv = acc[v];
        const int m = v + 8 * half;
        if (m < N_TOT) {
            if (col == 0)      atomicAdd(&s_dot_a[m], cv);
            else if (col == 1) atomicAdd(&s_dot_m[m], cv);
        }
    }
    __syncthreads();

    // ---- Softmax over the block axis (9 values); lane 0 -> attn, lane 1 -> mlp
    if (tid < 2) {
        const float* dots = (tid == 0) ? s_dot_a : s_dot_m;
        float*       ww   = (tid == 0) ? s_wa   : s_wm;
        float l[N_TOT];
        float mx = -3.402823466e38f;
        for (int n = 0; n < N_TOT; ++n) {
            const float rms = rsqrtf(s_ss[n] * (1.0f / (float)D_DIM) + EPS);
            l[n] = rms * dots[n] * (1.0f / 32.0f);   // /sqrt(1024)
            mx = fmaxf(mx, l[n]);
        }
        float sum = 0.0f;
        for (int n = 0; n < N_TOT; ++n) { float e = __expf(l[n] - mx); ww[n] = e; sum += e; }
        const float inv = 1.0f / sum;
        for (int n = 0; n < N_TOT; ++n) ww[n] *= inv;
    }
    __syncthreads();

    // ---- Weighted sums over n, 4 d's per thread, read LDS rows as float4
    const int d0 = tid * 4;
    float4 oa = make_float4(0.f, 0.f, 0.f, 0.f);
    float4 om = make_float4(0.f, 0.f, 0.f, 0.f);
#pragma unroll
    for (int n = 0; n < N_TOT; ++n) {
        const float wa = s_wa[n], wm = s_wm[n];
        const float4 xv = *(const float4*)(&xs[n * ROW_STRIDE + d0]);
        oa.x = fmaf(wa, xv.x, oa.x); oa.y = fmaf(wa, xv.y, oa.y);
        oa.z = fmaf(wa, xv.z, oa.z); oa.w = fmaf(wa, xv.w, oa.w);
        om.x = fmaf(wm, xv.x, om.x); om.y = fmaf(wm, xv.y, om.y);
        om.z = fmaf(wm, xv.z, om.z); om.w = fmaf(wm, xv.w, om.w);
    }
    const size_t obase = ((size_t)b * T_DIM + t) * (size_t)D_DIM + d0;
    *(float4*)(out + obase) = oa;                                            // h_attn
    *(float4*)(out + (size_t)B_DIM * T_DIM * D_DIM + obase) = om;            // h_mlp
}

extern "C" void kernel_launch(void* const* d_in, const int* in_sizes, int n_in,
                              void* d_out, int out_size, void* d_ws, size_t ws_size,
                              hipStream_t stream) {
    const float* blocks  = (const float*)d_in[0];
    const float* partial = (const float*)d_in[1];
    const float* pqa     = (const float*)d_in[2];
    const float* pqm     = (const float*)d_in[3];
    const float* nwa     = (const float*)d_in[4];
    const float* nwm     = (const float*)d_in[5];
    float* out = (float*)d_out;
    (void)in_sizes; (void)n_in; (void)out_size; (void)d_ws; (void)ws_size;

    hipLaunchKernelGGL(block_attn_kernel,
                       dim3(B_DIM * T_DIM), dim3(256), 0, stream,
                       blocks, partial, pqa, pqm, nwa, nwm, out);
}